// GPTOSSAttention_77704548319528
// MI455X (gfx1250) — compile-verified
//
#include <hip/hip_runtime.h>
#include <hip/hip_bf16.h>
#include <math.h>
#include <stdint.h>

// ---------------- problem constants ----------------
#define T_LEN   1536
#define HID     2880
#define NH      64
#define NKV     8
#define HD      64
#define WIN     128
#define QSZ     (NH*HD)            // 4096
#define KVSZ    (NKV*HD)           // 512
#define QKV_N   (QSZ + 2*KVSZ)     // 5120
#define SCALE_F 0.125f             // 1/sqrt(64)

#define A_STRIDE 20                // floats; 80B row = 16B-aligned, conflict-free
#define NSTAGE   3                 // async pipeline depth

typedef __attribute__((ext_vector_type(2))) float v2f;
typedef __attribute__((ext_vector_type(8))) float v8f;

// CDNA5 f32 matrix op: D(16x16) = A(16x4) * B(4x16) + C
__device__ __forceinline__ v8f wmma_f32(v2f a, v2f b, v8f c) {
  return __builtin_amdgcn_wmma_f32_16x16x4_f32(false, a, false, b, (short)0, c,
                                               false, false);
}

// CDNA5 async global->LDS copy, 16B per lane, tracked by ASYNCcnt.
// GVS addressing: saddr(64b SGPR pair) + vaddr(32b per-lane byte offset).
__device__ __forceinline__ void async_b128(const float* lds_dst,
                                           uint64_t gbase,
                                           uint32_t gofs_bytes) {
  uint32_t lofs = (uint32_t)(uintptr_t)lds_dst;   // low 32b of shared aperture
  asm volatile("global_load_async_to_lds_b128 %0, %1, %2"
               :: "v"(lofs), "v"(gofs_bytes), "s"(gbase)
               : "memory");
}

// ============================================================
// Tiled GEMM + bias:  C[M,N] = A[M,K] @ W[K,N] + bias
// 64x64 tile per 128-thread block (4 waves), K-chunks of 16.
// Triple-buffered LDS fed by async-to-LDS copies; steady-state
// loop body is branch-free (last two chunks peeled).
// Requires M%64==0, N%64==0, K%16==0, K/16>=3.
// ============================================================
struct GemmFrag {
  int mrow, koff, ncol;
};

__device__ __forceinline__ void compute_chunk(const float* Asb,
                                              const float* Bsb,
                                              v8f acc[4], GemmFrag f) {
  #pragma unroll
  for (int kk = 0; kk < 4; ++kk) {
    const int kb = kk * 4 + f.koff;
    v2f b;
    b.x = Bsb[(kb + 0) * 64 + f.ncol];
    b.y = Bsb[(kb + 1) * 64 + f.ncol];
    #pragma unroll
    for (int ms = 0; ms < 4; ++ms) {
      v2f a;
      a.x = Asb[(ms * 16 + f.mrow) * A_STRIDE + kb + 0];
      a.y = Asb[(ms * 16 + f.mrow) * A_STRIDE + kb + 1];
      acc[ms] = wmma_f32(a, b, acc[ms]);
    }
  }
}

__global__ __launch_bounds__(128)
void gemm_bias_kernel(const float* __restrict__ A, const float* __restrict__ W,
                      const float* __restrict__ bias, float* __restrict__ C,
                      int M, int N, int K) {
  __shared__ float As[NSTAGE][64 * A_STRIDE];
  __shared__ float Bs[NSTAGE][16 * 64];

  const int tid  = threadIdx.x;
  const int lane = tid & 31;
  const int wave = tid >> 5;
  const int m0 = blockIdx.y * 64;
  const int n0 = blockIdx.x * 64;
  GemmFrag frag;
  frag.mrow = lane & 15;                  // A-frag row (M), B-frag col (N)
  frag.koff = (lane >> 4) << 1;           // 0 for lanes 0-15, 2 for 16-31
  frag.ncol = (wave << 4) + frag.mrow;    // column within 64-wide tile
  const int halfsel = lane >> 4;

  // per-thread piece of a tile stage: 2 x 16B for A, 2 x 16B for B
  const int arow0 = tid >> 2,          acol0 = (tid & 3) << 2;
  const int arow1 = (tid + 128) >> 2,  acol1 = ((tid + 128) & 3) << 2;
  const int brow0 = tid >> 4,          bcol0 = (tid & 15) << 2;
  const int brow1 = (tid + 128) >> 4,  bcol1 = ((tid + 128) & 15) << 2;

  const uint64_t Abase = (uint64_t)(uintptr_t)(A + (size_t)m0 * K);
  const uint64_t Wbase = (uint64_t)(uintptr_t)(W + n0);

  // stage K-chunk `kc` (16 wide) into LDS stage `sb`: 4 async ops per thread
  auto stage = [&](int sb, int kc) {
    const int k0 = kc << 4;
    async_b128(&As[sb][arow0 * A_STRIDE + acol0], Abase,
               (uint32_t)(arow0 * K + k0 + acol0) * 4u);
    async_b128(&As[sb][arow1 * A_STRIDE + acol1], Abase,
               (uint32_t)(arow1 * K + k0 + acol1) * 4u);
    async_b128(&Bs[sb][brow0 * 64 + bcol0], Wbase,
               (uint32_t)((k0 + brow0) * N + bcol0) * 4u);
    async_b128(&Bs[sb][brow1 * 64 + bcol1], Wbase,
               (uint32_t)((k0 + brow1) * N + bcol1) * 4u);
  };

  v8f acc[4];
  #pragma unroll
  for (int ms = 0; ms < 4; ++ms)
    #pragma unroll
    for (int g = 0; g < 8; ++g) acc[ms][g] = 0.0f;

  const int nc = K >> 4;          // number of 16-wide K-chunks (>= 3)
  stage(0, 0);
  stage(1, 1);

  int buf = 0;
  // steady state: branch-free body, prefetch distance 2
  for (int ic = 0; ic < nc - 2; ++ic) {
    int pb = buf + 2; if (pb >= NSTAGE) pb -= NSTAGE;
    stage(pb, ic + 2);
    // async ops complete in order per wave: <=8 outstanding
    // => the 4 ops of the current chunk have landed
    asm volatile("s_wait_asynccnt 0x8" ::: "memory");
    __syncthreads();
    compute_chunk(As[buf], Bs[buf], acc, frag);
    __syncthreads();
    buf = (buf + 1 == NSTAGE) ? 0 : buf + 1;
  }
  // chunk nc-2
  asm volatile("s_wait_asynccnt 0x4" ::: "memory");
  __syncthreads();
  compute_chunk(As[buf], Bs[buf], acc, frag);
  __syncthreads();
  buf = (buf + 1 == NSTAGE) ? 0 : buf + 1;
  // chunk nc-1
  asm volatile("s_wait_asynccnt 0x0" ::: "memory");
  __syncthreads();
  compute_chunk(As[buf], Bs[buf], acc, frag);

  const int   ng = n0 + frag.ncol;
  const float bv = bias[ng];
  #pragma unroll
  for (int ms = 0; ms < 4; ++ms)
    #pragma unroll
    for (int g = 0; g < 8; ++g) {
      int row = m0 + ms * 16 + g + (halfsel << 3);
      C[(size_t)row * N + ng] = acc[ms][g] + bv;
    }
  (void)M;
}

// ============================================================
// YaRN RoPE applied in-place to Q and K inside the qkv buffer.
// ============================================================
__global__ void rope_kernel(const int* __restrict__ positions,
                            float* __restrict__ qkv) {
  int gid = blockIdx.x * blockDim.x + threadIdx.x;
  const int total = T_LEN * (NH + NKV) * 32;
  if (gid >= total) return;
  int d    = gid & 31;
  int rem  = gid >> 5;
  int head = rem % (NH + NKV);
  int t    = rem / (NH + NKV);

  size_t base = (size_t)t * QKV_N +
                (head < NH ? head * HD : QSZ + (head - NH) * HD);
  float x1 = qkv[base + d];
  float x2 = qkv[base + d + 32];

  const float half   = 32.0f;
  const float lnbase = logf(150000.0f);
  const float twoPi  = 6.28318530717958647692f;
  float freq  = powf(150000.0f, (float)d / half);
  float conc  = 0.1f * logf(32.0f) + 1.0f;
  float low   = half * logf(4096.0f / (32.0f * twoPi)) / lnbase;
  float high  = half * logf(4096.0f / (1.0f  * twoPi)) / lnbase;
  float ramp  = ((float)d - low) / (high - low);
  ramp        = fminf(fmaxf(ramp, 0.0f), 1.0f);
  float inv_freq = (1.0f / (32.0f * freq)) * ramp +
                   (1.0f / freq) * (1.0f - ramp);
  float ang = (float)positions[t] * inv_freq;
  float c = cosf(ang) * conc;
  float s = sinf(ang) * conc;

  qkv[base + d]      = x1 * c - x2 * s;
  qkv[base + d + 32] = x2 * c + x1 * s;
}

// ============================================================
// Sliding-window attention with sinks. One wave per
// (16-query block, head). Online softmax; S and P*V on WMMA.
// ============================================================
__global__ __launch_bounds__(32)
void attn_kernel(const float* __restrict__ qkv, const float* __restrict__ sinks,
                 float* __restrict__ attn) {
  const int qb   = blockIdx.x;        // 0..T/16-1
  const int h    = blockIdx.y;        // 0..63
  const int kvh  = h >> 3;            // GQA: 8 q-heads per kv-head
  const int lane = threadIdx.x;
  const int mrow    = lane & 15;
  const int koff    = (lane >> 4) << 1;
  const int halfsel = lane >> 4;

  __shared__ float Pt[16 * 17];

  const int qi = qb * 16 + mrow;
  const float* qrow = qkv + (size_t)qi * QKV_N + h * HD;
  float qreg[32];
  #pragma unroll
  for (int kb = 0; kb < 16; ++kb) {
    float2 qv = *reinterpret_cast<const float2*>(&qrow[kb * 4 + koff]);
    qreg[2 * kb]     = qv.x;
    qreg[2 * kb + 1] = qv.y;
  }

  float Mg[8], Lg[8];
  v8f oacc[4];
  #pragma unroll
  for (int g = 0; g < 8; ++g) { Mg[g] = -__builtin_inff(); Lg[g] = 0.0f; }
  #pragma unroll
  for (int dt = 0; dt < 4; ++dt)
    #pragma unroll
    for (int g = 0; g < 8; ++g) oacc[dt][g] = 0.0f;

  int jt0 = qb - 8; if (jt0 < 0) jt0 = 0;
  for (int jt = jt0; jt <= qb; ++jt) {
    // ---- S = Q K^T over HEAD_DIM=64 (16 WMMAs) ----
    v8f s;
    #pragma unroll
    for (int g = 0; g < 8; ++g) s[g] = 0.0f;
    const float* krow =
        qkv + (size_t)(jt * 16 + mrow) * QKV_N + QSZ + kvh * HD;
    #pragma unroll
    for (int kb = 0; kb < 16; ++kb) {
      float2 kv2 = *reinterpret_cast<const float2*>(&krow[kb * 4 + koff]);
      v2f a; a.x = qreg[2 * kb]; a.y = qreg[2 * kb + 1];
      v2f b; b.x = kv2.x;        b.y = kv2.y;
      s = wmma_f32(a, b, s);
    }

    // ---- mask + online softmax (rows replicated over 16-lane halves) ----
    float p[8];
    #pragma unroll
    for (int g = 0; g < 8; ++g) {
      int   i  = qb * 16 + g + (halfsel << 3);
      int   j  = jt * 16 + mrow;
      float sv = s[g] * SCALE_F;
      bool  ok = (j <= i) && (i - j < WIN);
      sv = ok ? sv : -__builtin_inff();
      float rm = sv;
      #pragma unroll
      for (int off = 8; off >= 1; off >>= 1)
        rm = fmaxf(rm, __shfl_xor(rm, off, 32));
      float mn = fmaxf(Mg[g], rm);
      float alpha, pv;
      if (mn == -__builtin_inff()) { alpha = 1.0f; pv = 0.0f; }
      else { alpha = expf(Mg[g] - mn); pv = expf(sv - mn); }
      float rs = pv;
      #pragma unroll
      for (int off = 8; off >= 1; off >>= 1)
        rs += __shfl_xor(rs, off, 32);
      Lg[g] = Lg[g] * alpha + rs;
      Mg[g] = mn;
      p[g]  = pv;
      #pragma unroll
      for (int dt = 0; dt < 4; ++dt) oacc[dt][g] *= alpha;
    }

    // ---- stage P through LDS to reshape C-layout -> A-fragments ----
    __syncthreads();
    #pragma unroll
    for (int g = 0; g < 8; ++g)
      Pt[(g + (halfsel << 3)) * 17 + mrow] = p[g];
    __syncthreads();

    // ---- O += P V (16 WMMAs: 4 k-steps x 4 d-tiles) ----
    #pragma unroll
    for (int kk = 0; kk < 4; ++kk) {
      v2f a;
      a.x = Pt[mrow * 17 + kk * 4 + koff + 0];
      a.y = Pt[mrow * 17 + kk * 4 + koff + 1];
      int r0 = jt * 16 + kk * 4 + koff;
      #pragma unroll
      for (int dt = 0; dt < 4; ++dt) {
        int d = dt * 16 + mrow;
        v2f b;
        b.x = qkv[(size_t)(r0 + 0) * QKV_N + QSZ + KVSZ + kvh * HD + d];
        b.y = qkv[(size_t)(r0 + 1) * QKV_N + QSZ + KVSZ + kvh * HD + d];
        oacc[dt] = wmma_f32(a, b, oacc[dt]);
      }
    }
    __syncthreads();   // Pt reused next tile
  }

  // ---- fold in sink logit, normalize, write ----
  const float snk = sinks[h];
  float osc[8];
  #pragma unroll
  for (int g = 0; g < 8; ++g) {
    float mf    = fmaxf(Mg[g], snk);
    float af    = expf(Mg[g] - mf);   // Mg finite: diagonal key always allowed
    float denom = Lg[g] * af + expf(snk - mf);
    osc[g] = af / denom;
  }
  #pragma unroll
  for (int dt = 0; dt < 4; ++dt)
    #pragma unroll
    for (int g = 0; g < 8; ++g) {
      int row = qb * 16 + g + (halfsel << 3);
      int col = h * HD + dt * 16 + mrow;
      attn[(size_t)row * QSZ + col] = oacc[dt][g] * osc[g];
    }
}

// ============================================================
extern "C" void kernel_launch(void* const* d_in, const int* in_sizes, int n_in,
                              void* d_out, int out_size, void* d_ws,
                              size_t ws_size, hipStream_t stream) {
  (void)in_sizes; (void)n_in; (void)out_size; (void)ws_size;
  const int*   positions = (const int*)d_in[0];
  const float* hidden    = (const float*)d_in[1];
  const float* W_qkv     = (const float*)d_in[2];
  const float* b_qkv     = (const float*)d_in[3];
  const float* W_o       = (const float*)d_in[4];
  const float* b_o       = (const float*)d_in[5];
  const float* sinks     = (const float*)d_in[6];
  float* out  = (float*)d_out;
  float* qkv  = (float*)d_ws;                         // T x 5120
  float* attn = qkv + (size_t)T_LEN * QKV_N;          // T x 4096

  // 1) QKV projection (1536x2880 @ 2880x5120 + bias)
  gemm_bias_kernel<<<dim3(QKV_N / 64, T_LEN / 64), 128, 0, stream>>>(
      hidden, W_qkv, b_qkv, qkv, T_LEN, QKV_N, HID);

  // 2) YaRN RoPE on Q and K, in place
  {
    int total = T_LEN * (NH + NKV) * 32;
    rope_kernel<<<(total + 255) / 256, 256, 0, stream>>>(positions, qkv);
  }

  // 3) sliding-window attention with sinks
  attn_kernel<<<dim3(T_LEN / 16, NH), 32, 0, stream>>>(qkv, sinks, attn);

  // 4) output projection (1536x4096 @ 4096x2880 + bias)
  gemm_bias_kernel<<<dim3(HID / 64, T_LEN / 64), 128, 0, stream>>>(
      attn, W_o, b_o, out, T_LEN, HID, QSZ);
}